// KGCN_51238959841862
// MI455X (gfx1250) — compile-verified
//
#include <hip/hip_runtime.h>
#include <math.h>

#define DIMN 64
#define NNB  16
#define NRELS 32
#define XS   68   // padded row stride (floats) for WMMA A tiles: 272B = 17*16, float4-aligned, bank-spread
#define WSP  66   // padded row stride for staged W: spreads K and K+2 rows across banks

typedef float v2f __attribute__((ext_vector_type(2)));
typedef float v8f __attribute__((ext_vector_type(8)));

__device__ __forceinline__ float sigm(float x) { return 1.0f / (1.0f + __expf(-x)); }

__global__ __launch_bounds__(256)
void kgcn_fused(const float* __restrict__ user_emb,
                const float* __restrict__ ent_emb,
                const float* __restrict__ rel_emb,
                const float* __restrict__ Wg,
                const float* __restrict__ bias,
                const int* __restrict__ u,
                const int* __restrict__ v,
                const int* __restrict__ adj_ent,
                const int* __restrict__ adj_rel,
                float* __restrict__ out)
{
    __shared__ float sW[DIMN][WSP];      // 16.5 KB staged weight
    __shared__ float sX[32][XS];         // GEMM input tile
    __shared__ float sY[32][XS];         // GEMM output tile (iter0 results)
    __shared__ float sUser[DIMN];
    __shared__ float sUr[NRELS];         // user . rel_emb[r]  (kills all rel gathers)
    __shared__ float sE0v[DIMN];
    __shared__ float sE1v[NNB][DIMN];
    __shared__ float sS1[NNB];
    __shared__ float sS2[NNB][NNB];
    __shared__ int   sE1[NNB];
    __shared__ int   sR1[NNB];
    __shared__ int   sE2[NNB][NNB];
    __shared__ int   sR2[NNB][NNB];
    __shared__ float sItem[DIMN];

    const int b    = blockIdx.x;
    const int t    = threadIdx.x;   // 0..255
    const int lane = t & 31;
    const int wave = t >> 5;

    const int e0  = v[b];
    const int uid = u[b];

    // ---------- Phase A: stage W, user vec, hop-0 indices, e0 embedding ----------
    for (int i = t; i < DIMN * DIMN; i += 256) sW[i >> 6][i & 63] = Wg[i];
    if (t < DIMN)       sUser[t]     = user_emb[(size_t)uid * DIMN + t];
    else if (t < 128)   sE0v[t - 64] = ent_emb[(size_t)e0 * DIMN + (t - 64)];
    else if (t < 144)   sE1[t - 128] = adj_ent[(size_t)e0 * NNB + (t - 128)];
    else if (t < 160)   sR1[t - 144] = adj_rel[(size_t)e0 * NNB + (t - 144)];
    __syncthreads();

    // ---------- Phase B: hop-1 adjacency, e1 rows, ur[] ----------
    {
        const int j = t >> 4, k = t & 15;
        const int ej = sE1[j];
        sE2[j][k] = adj_ent[(size_t)ej * NNB + k];
        sR2[j][k] = adj_rel[(size_t)ej * NNB + k];
        const int c4 = k * 4;
        const float4 p = *reinterpret_cast<const float4*>(&ent_emb[(size_t)ej * DIMN + c4]);
        *reinterpret_cast<float4*>(&sE1v[j][c4]) = p;
    }
    if (t < NRELS) {
        const float* rp = &rel_emb[t * DIMN];
        float acc = 0.f;
        #pragma unroll
        for (int d = 0; d < DIMN; ++d) acc += sUser[d] * rp[d];
        sUr[t] = acc;
    }
    __syncthreads();

    // ---------- Phase C: softmax scores (pure LDS) ----------
    if (t < NNB) {
        float l[NNB], m = -1e30f;
        #pragma unroll
        for (int k = 0; k < NNB; ++k) { l[k] = sUr[sR2[t][k]]; m = fmaxf(m, l[k]); }
        float s = 0.f;
        #pragma unroll
        for (int k = 0; k < NNB; ++k) { l[k] = __expf(l[k] - m); s += l[k]; }
        const float inv = 1.f / s;
        #pragma unroll
        for (int k = 0; k < NNB; ++k) sS2[t][k] = l[k] * inv;
    } else if (t == 16) {
        float l[NNB], m = -1e30f;
        #pragma unroll
        for (int k = 0; k < NNB; ++k) { l[k] = sUr[sR1[k]]; m = fmaxf(m, l[k]); }
        float s = 0.f;
        #pragma unroll
        for (int k = 0; k < NNB; ++k) { l[k] = __expf(l[k] - m); s += l[k]; }
        const float inv = 1.f / s;
        #pragma unroll
        for (int k = 0; k < NNB; ++k) sS1[k] = l[k] * inv;
    }
    __syncthreads();

    // ---------- Phase D: THE gather (HBM-bound): weighted sum of 256 ent rows ----------
    {
        const int j = t >> 4, c4 = (t & 15) * 4;  // 16 threads cover one 256B row: coalesced
        float4 acc = make_float4(0.f, 0.f, 0.f, 0.f);
        #pragma unroll
        for (int k = 0; k < NNB; ++k) {
            const float w = sS2[j][k];
            const float4 p = *reinterpret_cast<const float4*>(
                &ent_emb[(size_t)sE2[j][k] * DIMN + c4]);
            acc.x += w * p.x; acc.y += w * p.y; acc.z += w * p.z; acc.w += w * p.w;
        }
        float4 xo;
        xo.x = sE1v[j][c4 + 0] + acc.x;
        xo.y = sE1v[j][c4 + 1] + acc.y;
        xo.z = sE1v[j][c4 + 2] + acc.z;
        xo.w = sE1v[j][c4 + 3] + acc.w;
        *reinterpret_cast<float4*>(&sX[j][c4]) = xo;
    }
    if (t < DIMN) {   // hop-0 aggregation from LDS-resident e1 rows -> X row 16
        float acc = 0.f;
        #pragma unroll
        for (int k = 0; k < NNB; ++k) acc += sS1[k] * sE1v[k][t];
        sX[16][t] = sE0v[t] + acc;
    }
    __syncthreads();

    // ---------- Phase E: GEMM1 via V_WMMA_F32_16X16X4_F32 : Y = sigmoid(X(32x64) @ W + b) ----------
    {
        const int mtile = wave >> 2, ntile = wave & 3;
        const int arow  = (mtile << 4) + (lane & 15);
        const int koff  = (lane >> 4) << 1;              // 0 (lanes 0-15) or 2 (lanes 16-31)
        const int bcol  = (ntile << 4) + (lane & 15);
        v8f c = {0.f, 0.f, 0.f, 0.f, 0.f, 0.f, 0.f, 0.f};
        #pragma unroll
        for (int kk = 0; kk < 16; ++kk) {
            const int kb = (kk << 2) + koff;
            v2f a;  a.x  = sX[arow][kb];   a.y  = sX[arow][kb + 1];   // A 16x4 layout
            v2f bb; bb.x = sW[kb][bcol];   bb.y = sW[kb + 1][bcol];   // B 4x16 layout
            c = __builtin_amdgcn_wmma_f32_16x16x4_f32(false, a, false, bb,
                                                      (short)0, c, false, false);
        }
        const float bv = bias[bcol];
        const int rbase = (mtile << 4) + ((lane >> 4) << 3);
        #pragma unroll
        for (int g = 0; g < 8; ++g)
            sY[rbase + g][bcol] = sigm(c[g] + bv);   // iter0 activation = sigmoid
    }
    __syncthreads();

    // ---------- Phase F: iter1 aggregation -> X row 0 ----------
    if (t < DIMN) {
        float acc = sY[16][t];                 // hop-0 self vector
        #pragma unroll
        for (int k = 0; k < NNB; ++k) acc += sS1[k] * sY[k][t];
        sX[0][t] = acc;
    }
    __syncthreads();

    // ---------- Phase G: GEMM2 (1 valid row) : item = tanh(row0 @ W + b) ----------
    if (wave < 4) {
        const int ntile = wave;
        const int arow  = lane & 15;
        const int koff  = (lane >> 4) << 1;
        const int bcol  = (ntile << 4) + (lane & 15);
        v8f c = {0.f, 0.f, 0.f, 0.f, 0.f, 0.f, 0.f, 0.f};
        #pragma unroll
        for (int kk = 0; kk < 16; ++kk) {
            const int kb = (kk << 2) + koff;
            v2f a;  a.x  = sX[arow][kb];   a.y  = sX[arow][kb + 1];
            v2f bb; bb.x = sW[kb][bcol];   bb.y = sW[kb + 1][bcol];
            c = __builtin_amdgcn_wmma_f32_16x16x4_f32(false, a, false, bb,
                                                      (short)0, c, false, false);
        }
        if (lane < 16)   // VGPR0, lanes 0-15 hold M=0 (the only valid row)
            sItem[bcol] = tanhf(c[0] + bias[bcol]);
    }
    __syncthreads();

    // ---------- Phase H: score = sigmoid(user . item) ----------
    if (t < 32) {
        float p = sUser[t] * sItem[t] + sUser[t + 32] * sItem[t + 32];
        #pragma unroll
        for (int off = 16; off > 0; off >>= 1) p += __shfl_xor(p, off, 32);
        if (t == 0) out[b] = sigm(p);
    }
}

extern "C" void kernel_launch(void* const* d_in, const int* in_sizes, int n_in,
                              void* d_out, int out_size, void* d_ws, size_t ws_size,
                              hipStream_t stream) {
    const float* user_emb = (const float*)d_in[0];
    const float* ent_emb  = (const float*)d_in[1];
    const float* rel_emb  = (const float*)d_in[2];
    const float* W        = (const float*)d_in[3];
    const float* bias     = (const float*)d_in[4];
    const int*   u        = (const int*)d_in[5];
    const int*   v        = (const int*)d_in[6];
    const int*   adj_ent  = (const int*)d_in[7];
    const int*   adj_rel  = (const int*)d_in[8];
    float* out = (float*)d_out;
    const int B = in_sizes[5];   // BATCH from u

    kgcn_fused<<<B, 256, 0, stream>>>(user_emb, ent_emb, rel_emb, W, bias,
                                      u, v, adj_ent, adj_rel, out);
}